// NoisyTopkRouter_22814866276627
// MI455X (gfx1250) — compile-verified
//
#include <hip/hip_runtime.h>
#include <hip/hip_bf16.h>
#include <math.h>

// NoisyTopkRouter for MI455X (gfx1250), wave32.
//
// gate = H @ Wg^T + bg ; noiseproj = H @ Wn^T + bn    (N=131072, D=1024, E=8)
// noisy = gate + noise * softplus(noiseproj)
// top-2 of softmax(noisy) renormalized.
//
// Both projections fused into a single V_WMMA_F32_16X16X4_F32 chain with a
// combined 16-column B matrix (cols 0-7 = gate experts, cols 8-15 = noise
// experts). f32 WMMA keeps reference-precision accumulation; the kernel is
// HBM-bound on the 512 MB read of H (~22 us floor at 23.3 TB/s), and
// 8 flop/byte intensity is within reach of the f32 matrix pipe.

typedef __attribute__((ext_vector_type(2))) float v2f;
typedef __attribute__((ext_vector_type(8))) float v8f;

#define EMBED_D 1024
#define NEXP    8

__global__ __launch_bounds__(256) void noisy_topk_router_kernel(
    const float* __restrict__ H,      // [N, 1024]
    const float* __restrict__ noise,  // [N, 8]
    const float* __restrict__ Wg,     // [8, 1024]
    const float* __restrict__ bg,     // [8]
    const float* __restrict__ Wn,     // [8, 1024]
    const float* __restrict__ bn,     // [8]
    float* __restrict__ out,
    int nTokens)
{
    // wave-private 16x16 f32 tile staging (8 waves per block)
    __shared__ float tileLds[8 * 16 * 16];

    const int lane = threadIdx.x & 31;
    const int wave = threadIdx.x >> 5;
    const int col  = lane & 15;   // combined output column (0-7 gate, 8-15 noise)
    const int hi   = lane >> 4;   // 0 -> K+0/+1 ; 1 -> K+2/+3 (A & B f32 layouts)

    // per-lane pointer to the weight row feeding combined column `col`
    const float* wrow = (col < NEXP) ? (Wg + (size_t)col * EMBED_D)
                                     : (Wn + (size_t)(col - NEXP) * EMBED_D);

    const int nTiles = nTokens >> 4;
    const int tileStride = gridDim.x * 8;

    // output layout (flat, in reference return order)
    float* outW     = out;                               // topk_weights  [N,2]
    int*   outIdx   = (int*)(out + 2 * (size_t)nTokens); // selected      [N,2]
    float* outNoisy = out + 4 * (size_t)nTokens;         // noisy_logits  [N,8]
    float* outGate  = outNoisy + 8 * (size_t)nTokens;    // gate_logits   [N,8]

    float* myTile = tileLds + wave * 256;

    for (int tile = blockIdx.x * 8 + wave; tile < nTiles; tile += tileStride) {
        // A fragment source: row of H owned by this lane (lanes L and L+16
        // share row L&15, differing only in the K sub-offset `hi*2`)
        const float* hrow = H + ((size_t)tile * 16 + (size_t)col) * EMBED_D + 2 * hi;
        const float* brow = wrow + 2 * hi;

        v8f acc = {0.f, 0.f, 0.f, 0.f, 0.f, 0.f, 0.f, 0.f};

        #pragma unroll 8
        for (int k = 0; k < EMBED_D; k += 4) {
            v2f a = *(const v2f*)(hrow + k);  // global_load_b64
            v2f b = *(const v2f*)(brow + k);  // global_load_b64 (cache-hot)
            acc = __builtin_amdgcn_wmma_f32_16x16x4_f32(
                /*neg_a=*/false, a, /*neg_b=*/false, b,
                /*c_mod=*/(short)0, acc, /*reuse_a=*/false, /*reuse_b=*/false);
        }

        // Scatter D tile to LDS. C/D layout: VGPR i -> row (i + 8*hi), col = lane&15.
        #pragma unroll
        for (int i = 0; i < 8; ++i) {
            myTile[(i + 8 * hi) * 16 + col] = acc[i];
        }
        // DS ops are in-order per wave; wait for this wave's stores so the
        // cross-lane reads below observe them (tile region is wave-private,
        // so no workgroup barrier is required).
        asm volatile("s_wait_dscnt 0x0" ::: "memory");

        if (lane < 16) {
            const size_t t = (size_t)tile * 16 + (size_t)lane;
            const float* row = myTile + lane * 16;

            float gate[NEXP], noisy[NEXP];
            #pragma unroll
            for (int e = 0; e < NEXP; ++e) {
                float g  = row[e] + bg[e];
                float nl = row[NEXP + e] + bn[e];
                // stable softplus == jax logaddexp(x, 0)
                float sp = fmaxf(nl, 0.f) + log1pf(expf(-fabsf(nl)));
                gate[e]  = g;
                noisy[e] = fmaf(noise[t * NEXP + e], sp, g);
            }

            // top-2 over noisy logits (softmax is monotonic; ties -> lowest index,
            // matching jax.lax.top_k's stable ordering)
            int i1 = 0;
            #pragma unroll
            for (int e = 1; e < NEXP; ++e)
                if (noisy[e] > noisy[i1]) i1 = e;
            int i2 = (i1 == 0) ? 1 : 0;
            #pragma unroll
            for (int e = 0; e < NEXP; ++e)
                if (e != i1 && noisy[e] > noisy[i2]) i2 = e;

            // renormalized top-2 softmax weights: denominator cancels
            float d  = noisy[i2] - noisy[i1];   // <= 0
            float ex = expf(d);
            float w1 = 1.f / (1.f + ex);
            float w2 = 1.f - w1;

            outW[t * 2 + 0]   = w1;
            outW[t * 2 + 1]   = w2;
            outIdx[t * 2 + 0] = i1;
            outIdx[t * 2 + 1] = i2;
            #pragma unroll
            for (int e = 0; e < NEXP; ++e) {
                outNoisy[t * NEXP + e] = noisy[e];
                outGate[t * NEXP + e]  = gate[e];
            }
        }
    }
}

extern "C" void kernel_launch(void* const* d_in, const int* in_sizes, int n_in,
                              void* d_out, int out_size, void* d_ws, size_t ws_size,
                              hipStream_t stream) {
    const float* H     = (const float*)d_in[0];
    const float* noise = (const float*)d_in[1];
    const float* Wg    = (const float*)d_in[2];
    const float* bg    = (const float*)d_in[3];
    const float* Wn    = (const float*)d_in[4];
    const float* bn    = (const float*)d_in[5];

    const int nTokens = in_sizes[0] / EMBED_D;   // 131072
    const int nTiles  = nTokens / 16;            // 8192 (one wave each)
    const int blocks  = (nTiles + 7) / 8;        // 8 waves (256 thr) per block

    noisy_topk_router_kernel<<<blocks, 256, 0, stream>>>(
        H, noise, Wg, bg, Wn, bn, (float*)d_out, nTokens);
}